// SineRNN_66202625901259
// MI455X (gfx1250) — compile-verified
//
#include <hip/hip_runtime.h>

// Problem constants (from reference): B=256, T=4096, I=1, H=32
#define B_ 256
#define T_ 4096
#define H_ 32
#define LDS_STRIDE 20  // padded row stride (words) to avoid LDS bank conflicts

typedef float v2f __attribute__((ext_vector_type(2)));
typedef float v8f __attribute__((ext_vector_type(8)));

// tanh without libcalls: prefer hardware v_tanh_f32, else exp2/rcp formulation.
__device__ __forceinline__ float hw_tanhf(float v) {
#if __has_builtin(__builtin_amdgcn_tanhf)
    return __builtin_amdgcn_tanhf(v);
#else
    // tanh(x) = (e^2x - 1) / (e^2x + 1); clamp so exp2 stays finite.
    const float xc = __builtin_fminf(__builtin_fmaxf(v, -9.0f), 9.0f);
    const float t  = __builtin_amdgcn_exp2f(xc * 2.8853900817779268f); // 2*log2(e)
    const float r  = __builtin_amdgcn_rcpf(t + 1.0f);                  // v_rcp_f32
    return (t - 1.0f) * r;
#endif
}

// One wave (32 lanes) advances 16 batch chains through all T steps.
// Step math: D(c) [16 hidden_new x 16 batch] = W_hh[16c..16c+15][:] @ H^T, c=0,1
//   A operand  (16x4 f32): lane m (both halves) = W_hh row 16c+(lane&15),
//                          VGPR0/1 = K = 4j + 2*(lane>>4) + {0,1}
//   B operand  (4x16 f32): lane n holds H[4j + 2*(lane>>4) + {0,1}][lane&15]
//   C/D (16x16 f32): lane n = batch, VGPR r = hidden_new 8*(lane>>4)+r
__global__ __launch_bounds__(32, 1)
void SineRNN_wmma_kernel(const float* __restrict__ x,
                         const float* __restrict__ W_ih,
                         const float* __restrict__ W_hh,
                         const float* __restrict__ b_ih,
                         const float* __restrict__ b_hh,
                         const float* __restrict__ W_fc,
                         const float* __restrict__ b_fc,
                         float* __restrict__ out)
{
    __shared__ float hbuf[H_ * LDS_STRIDE];  // H[hidden][batch] repack buffer

    const int lane  = threadIdx.x & 31;
    const int n     = lane & 15;   // batch column owned by this lane
    const int hi    = lane >> 4;   // lane half
    const int bbase = blockIdx.x * 16;

    // ---- Preload A operands (W_hh tiles), constant over time: 32 VGPRs ----
    v2f A[2][8];
#pragma unroll
    for (int c = 0; c < 2; ++c) {
#pragma unroll
        for (int j = 0; j < 8; ++j) {
            const int row = 16 * c + n;
            const int k0  = 4 * j + 2 * hi;
            A[c][j].x = W_hh[row * H_ + k0];
            A[c][j].y = W_hh[row * H_ + k0 + 1];
        }
    }

    // ---- Per-lane constants in C/D layout: hidden index 16c + 8*hi + r ----
    float wih[2][8], bia[2][8], wfc[2][8];
#pragma unroll
    for (int c = 0; c < 2; ++c) {
#pragma unroll
        for (int r = 0; r < 8; ++r) {
            const int h = 16 * c + 8 * hi + r;
            wih[c][r] = W_ih[h];                 // W_ih is (H,1)
            bia[c][r] = b_ih[h] + b_hh[h];
            wfc[c][r] = W_fc[h];                 // W_fc is (1,H)
        }
    }
    const float bfc = b_fc[0];

    const float* xrow = x   + (size_t)(bbase + n) * T_;  // x[b][t], I==1
    float*       orow = out + (size_t)(bbase + n) * T_;  // out[b][t]

    // h0 = 0 -> B operand starts zero
    v2f Bop[8];
#pragma unroll
    for (int j = 0; j < 8; ++j) { Bop[j].x = 0.0f; Bop[j].y = 0.0f; }

    for (int t = 0; t < T_; ++t) {
        const float xv = xrow[t];
        __builtin_prefetch(xrow + t + 96, 0, 0);  // global_prefetch_b8

        // C init = xp_t = x*W_ih + (b_ih + b_hh), already in C/D layout
        v8f acc0, acc1;
#pragma unroll
        for (int r = 0; r < 8; ++r) {
            acc0[r] = __builtin_fmaf(xv, wih[0][r], bia[0][r]);
            acc1[r] = __builtin_fmaf(xv, wih[1][r], bia[1][r]);
        }

        // Two independent chains of 8 accumulating fp32 WMMAs (K=32 total)
#pragma unroll
        for (int j = 0; j < 8; ++j) {
            acc0 = __builtin_amdgcn_wmma_f32_16x16x4_f32(
                false, A[0][j], false, Bop[j], (short)0, acc0, false, false);
            acc1 = __builtin_amdgcn_wmma_f32_16x16x4_f32(
                false, A[1][j], false, Bop[j], (short)0, acc1, false, false);
        }

        // tanh + fc partial dot (per-lane covers 16 of the 32 hidden values)
        float s = 0.0f;
#pragma unroll
        for (int r = 0; r < 8; ++r) {
            const float h0 = hw_tanhf(acc0[r]);
            const float h1 = hw_tanhf(acc1[r]);
            acc0[r] = h0;
            acc1[r] = h1;
            s = __builtin_fmaf(wfc[0][r], h0, s);
            s = __builtin_fmaf(wfc[1][r], h1, s);
        }
        // combine the two lane-halves (lo half: hidden 0..7,16..23; hi half: 8..15,24..31)
        s += __shfl_xor(s, 16, 32);
        if (hi == 0) orow[t] = s + bfc;   // one store per batch per step

        // ---- Repack D (C/D layout) -> B operand layout via LDS ----
        // store: H[hidden = 16c + 8*hi + r][n]
#pragma unroll
        for (int r = 0; r < 8; ++r) {
            hbuf[(8 * hi + r)      * LDS_STRIDE + n] = acc0[r];
            hbuf[(16 + 8 * hi + r) * LDS_STRIDE + n] = acc1[r];
        }
        // load: B_j = H[4j + 2*hi + {0,1}][n]  (single wave owns hbuf; LDS is
        // in-order within a wave, compiler inserts s_wait_dscnt)
#pragma unroll
        for (int j = 0; j < 8; ++j) {
            const int k0 = 4 * j + 2 * hi;
            Bop[j].x = hbuf[k0       * LDS_STRIDE + n];
            Bop[j].y = hbuf[(k0 + 1) * LDS_STRIDE + n];
        }
    }
}

extern "C" void kernel_launch(void* const* d_in, const int* in_sizes, int n_in,
                              void* d_out, int out_size, void* d_ws, size_t ws_size,
                              hipStream_t stream) {
    const float* x    = (const float*)d_in[0];
    const float* W_ih = (const float*)d_in[1];
    const float* W_hh = (const float*)d_in[2];
    const float* b_ih = (const float*)d_in[3];
    const float* b_hh = (const float*)d_in[4];
    const float* W_fc = (const float*)d_in[5];
    const float* b_fc = (const float*)d_in[6];
    float* out = (float*)d_out;

    dim3 grid(B_ / 16);   // 16 blocks, one wave each, 16 batch chains per wave
    dim3 block(32);
    SineRNN_wmma_kernel<<<grid, block, 0, stream>>>(
        x, W_ih, W_hh, b_ih, b_hh, W_fc, b_fc, out);
}